// GraphSAGEModel_73521250173079
// MI455X (gfx1250) — compile-verified
//
#include <hip/hip_runtime.h>

typedef __attribute__((ext_vector_type(2))) float v2f;
typedef __attribute__((ext_vector_type(8))) float v8f;

#define N_NODES 100000
#define N_EDGES 1600000
#define DIM_IN  128
#define DIM_HID 128
#define DIM_OUT 64

// ---------------- zero fill (float4, grid-stride) ----------------
__global__ void k_zero4(float4* __restrict__ p, int n4) {
  int i = blockIdx.x * blockDim.x + threadIdx.x;
  int stride = gridDim.x * blockDim.x;
  float4 z; z.x = 0.f; z.y = 0.f; z.z = 0.f; z.w = 0.f;
  for (; i < n4; i += stride) p[i] = z;
}

// ---------------- degree / inverse degree ----------------
__global__ void k_degree(const int* __restrict__ dst, float* __restrict__ deg) {
  int e = blockIdx.x * blockDim.x + threadIdx.x;
  if (e < N_EDGES) atomicAdd(&deg[dst[e]], 1.0f);
}

__global__ void k_invdeg(float* __restrict__ deg) {
  int i = blockIdx.x * blockDim.x + threadIdx.x;
  if (i < N_NODES) {
    float d = deg[i];
    deg[i] = 1.0f / fmaxf(d, 1.0f);
  }
}

// ---------------- neighbor scatter-sum: one wave32 per edge ----------------
// feat dim = 128: each lane handles one float4 (32 lanes * 4 = 128)
__global__ void k_scatter128(const int* __restrict__ src, const int* __restrict__ dst,
                             const float* __restrict__ feat, float* __restrict__ agg) {
  int wave = (blockIdx.x * blockDim.x + threadIdx.x) >> 5;
  int lane = threadIdx.x & 31;
  if (wave >= N_EDGES) return;
  int s = src[wave];
  int d = dst[wave];
  const float4* frow = (const float4*)(feat + (size_t)s * 128);
  float4 v = frow[lane];
  float* orow = agg + (size_t)d * 128 + lane * 4;
  atomicAdd(orow + 0, v.x);
  atomicAdd(orow + 1, v.y);
  atomicAdd(orow + 2, v.z);
  atomicAdd(orow + 3, v.w);
}

// ---------------- fused SAGEConv GEMM via fp32 WMMA ----------------
// out[row, col] = (agg[row]*invdeg[row]) @ Wl + xin[row] @ Wr + bias  (optional relu)
// K = 128 for both layers. One wave computes one 16x16 output tile with
// V_WMMA_F32_16X16X4_F32; block = (NCOL/16) waves covering 16 rows x NCOL cols.
template <int NCOL, bool RELU>
__global__ void k_sage_gemm(const float* __restrict__ agg, const float* __restrict__ invdeg,
                            const float* __restrict__ xin, const float* __restrict__ Wl,
                            const float* __restrict__ Wr, const float* __restrict__ bias,
                            float* __restrict__ out) {
  const int K = 128;
  int wave = threadIdx.x >> 5;
  int lane = threadIdx.x & 31;
  int half = lane >> 4;   // 0: lanes 0-15, 1: lanes 16-31
  int l15  = lane & 15;

  int row0 = blockIdx.x * 16;
  int col0 = wave * 16;
  int col  = col0 + l15;

  // A-fragment row for this lane (both halves cover M=0..15)
  int am = row0 + l15;
  float id = invdeg[am];
  const float* aggRow = agg + (size_t)am * K;
  const float* xRow   = xin + (size_t)am * K;

  // C/D layout: VGPR r -> row r + half*8, col = col0 + l15. Bias depends on col only.
  v8f acc;
  float bv = bias[col];
#pragma unroll
  for (int r = 0; r < 8; ++r) acc[r] = bv;

  for (int kk = 0; kk < K; kk += 4) {
    int ka = kk + half * 2;  // A layout: lanes<16 hold K={kk,kk+1}, lanes>=16 hold K={kk+2,kk+3}
    v2f aL, aR, bL, bR;
    aL.x = aggRow[ka] * id;
    aL.y = aggRow[ka + 1] * id;
    aR.x = xRow[ka];
    aR.y = xRow[ka + 1];
    // B (4x16, row-striped across lanes): mirror of A's K placement
    bL.x = Wl[(size_t)(ka + 0) * NCOL + col];
    bL.y = Wl[(size_t)(ka + 1) * NCOL + col];
    bR.x = Wr[(size_t)(ka + 0) * NCOL + col];
    bR.y = Wr[(size_t)(ka + 1) * NCOL + col];
    acc = __builtin_amdgcn_wmma_f32_16x16x4_f32(false, aL, false, bL, (short)0, acc, false, false);
    acc = __builtin_amdgcn_wmma_f32_16x16x4_f32(false, aR, false, bR, (short)0, acc, false, false);
  }

#pragma unroll
  for (int r = 0; r < 8; ++r) {
    float v = acc[r];
    if (RELU) v = fmaxf(v, 0.f);
    out[(size_t)(row0 + r + half * 8) * NCOL + col] = v;
  }
}

// ---------------- in-place log-softmax over 64 cols: one wave32 per node ----------------
__global__ void k_logsoftmax64(float* __restrict__ logits) {
  int wave = (blockIdx.x * blockDim.x + threadIdx.x) >> 5;
  int lane = threadIdx.x & 31;
  if (wave >= N_NODES) return;
  float2* row = (float2*)(logits + (size_t)wave * DIM_OUT);
  float2 v = row[lane];  // cols 2*lane, 2*lane+1
  float m = fmaxf(v.x, v.y);
#pragma unroll
  for (int off = 16; off > 0; off >>= 1) m = fmaxf(m, __shfl_xor(m, off, 32));
  float e = __expf(v.x - m) + __expf(v.y - m);
#pragma unroll
  for (int off = 16; off > 0; off >>= 1) e += __shfl_xor(e, off, 32);
  float lse = m + __logf(e);
  float2 o; o.x = v.x - lse; o.y = v.y - lse;
  row[lane] = o;
}

// ---------------- launcher ----------------
extern "C" void kernel_launch(void* const* d_in, const int* in_sizes, int n_in,
                              void* d_out, int out_size, void* d_ws, size_t ws_size,
                              hipStream_t stream) {
  const float* x   = (const float*)d_in[0];
  const int*   ei  = (const int*)d_in[1];
  const float* Wl1 = (const float*)d_in[2];
  const float* Wr1 = (const float*)d_in[3];
  const float* b1  = (const float*)d_in[4];
  const float* Wl2 = (const float*)d_in[5];
  const float* Wr2 = (const float*)d_in[6];
  const float* b2  = (const float*)d_in[7];
  float* out = (float*)d_out;

  const int* src = ei;             // edge_index[0, :]
  const int* dst = ei + N_EDGES;   // edge_index[1, :]

  // workspace layout (floats)
  const size_t DEG_OFF = 0;
  const size_t DEG_PAD = 100352;                    // N rounded up, keeps agg 128B-aligned
  const size_t AGG_OFF = DEG_PAD;                   // N * 128
  const size_t H_OFF   = AGG_OFF + (size_t)N_NODES * DIM_HID;
  float* ws  = (float*)d_ws;
  float* deg = ws + DEG_OFF;   // becomes inv_deg in place
  float* agg = ws + AGG_OFF;
  float* h   = ws + H_OFF;

  const int ZB = 2048, ZT = 256;
  int span4 = (int)((DEG_PAD + (size_t)N_NODES * DIM_HID) / 4);
  int agg4  = (int)(((size_t)N_NODES * DIM_HID) / 4);

  // ---- phase 0: init deg + agg ----
  k_zero4<<<ZB, ZT, 0, stream>>>((float4*)deg, span4);

  // ---- degree / inv_deg ----
  k_degree<<<(N_EDGES + 255) / 256, 256, 0, stream>>>(dst, deg);
  k_invdeg<<<(N_NODES + 255) / 256, 256, 0, stream>>>(deg);

  // ---- layer 1 ----
  {
    size_t waves = (size_t)N_EDGES;
    int blocks = (int)((waves * 32 + 255) / 256);
    k_scatter128<<<blocks, 256, 0, stream>>>(src, dst, x, agg);
  }
  k_sage_gemm<DIM_HID, true><<<N_NODES / 16, (DIM_HID / 16) * 32, 0, stream>>>(
      agg, deg, x, Wl1, Wr1, b1, h);

  // ---- layer 2 ----
  k_zero4<<<ZB, ZT, 0, stream>>>((float4*)agg, agg4);
  {
    size_t waves = (size_t)N_EDGES;
    int blocks = (int)((waves * 32 + 255) / 256);
    k_scatter128<<<blocks, 256, 0, stream>>>(src, dst, h, agg);
  }
  k_sage_gemm<DIM_OUT, false><<<N_NODES / 16, (DIM_OUT / 16) * 32, 0, stream>>>(
      agg, deg, h, Wl2, Wr2, b2, out);

  // ---- log-softmax (in place on d_out) ----
  {
    int blocks = (int)(((size_t)N_NODES * 32 + 255) / 256);
    k_logsoftmax64<<<blocks, 256, 0, stream>>>(out);
  }
}